// HungarianMatcher_987842478748
// MI455X (gfx1250) — compile-verified
//
#include <hip/hip_runtime.h>

// Problem constants (B=32, N=300, C=92, T=50)
#define NROWS 9600   // B*N
#define NCOLS 1600   // B*T
#define NCLS  92

typedef float v2f __attribute__((ext_vector_type(2)));
typedef float v8f __attribute__((ext_vector_type(8)));

// ---------------- Kernel 1: row softmax (wave per row) ----------------
__global__ __launch_bounds__(256) void hm_softmax_rows(
    const float* __restrict__ logits, float* __restrict__ prob) {
  const int wave = threadIdx.x >> 5;
  const int lane = threadIdx.x & 31;
  const int row  = blockIdx.x * 8 + wave;
  const float* src = logits + (long)row * NCLS;

  float l0 = src[lane];
  float l1 = src[lane + 32];
  float l2 = (lane < NCLS - 64) ? src[lane + 64] : -3.402823466e38f;

  float m = fmaxf(fmaxf(l0, l1), l2);
  #pragma unroll
  for (int off = 16; off > 0; off >>= 1) m = fmaxf(m, __shfl_xor(m, off, 32));

  float e0 = __expf(l0 - m);
  float e1 = __expf(l1 - m);
  float e2 = (lane < NCLS - 64) ? __expf(l2 - m) : 0.0f;

  float s = e0 + e1 + e2;
  #pragma unroll
  for (int off = 16; off > 0; off >>= 1) s += __shfl_xor(s, off, 32);

  const float inv = 1.0f / s;
  float* dst = prob + (long)row * NCLS;
  dst[lane]      = e0 * inv;
  dst[lane + 32] = e1 * inv;
  if (lane < NCLS - 64) dst[lane + 64] = e2 * inv;
}

// ---------------- Kernel 2: cost matrix, WMMA class-cost ----------------
// Block: 128 threads (4 waves). Tile: 16 rows x 64 cols (each wave: 16x16).
// cost_class via f32 WMMA against an in-register one-hot B matrix (exact).
__global__ __launch_bounds__(128) void hm_cost_kernel(
    const float* __restrict__ prob,        // (9600, 92)
    const float* __restrict__ pred_boxes,  // (9600, 4) cxcywh
    const int*   __restrict__ tgt_labels,  // (1600,)
    const float* __restrict__ tgt_boxes,   // (1600, 4) cxcywh
    float* __restrict__ out) {             // (9600, 1600)
  __shared__ float probLds[16 * 97];       // padded stride 97 (conflict-free)
  __shared__ float rowBox[16][9];          // cx,cy,w,h,x0,y0,x1,y1,area
  __shared__ float colBox[64][9];
  __shared__ int   colId[64];

  const int rowBase = blockIdx.x * 16;
  const int colBase = blockIdx.y * 64;
  const int t = threadIdx.x;

  // Stage 16x92 prob slab (contiguous in global since rows are contiguous).
  for (int idx = t; idx < 16 * NCLS; idx += 128) {
    const int r = idx / NCLS;
    const int c = idx - r * NCLS;
    probLds[r * 97 + c] = prob[(long)rowBase * NCLS + idx];
  }
  // Stage query boxes + derived xyxy/area.
  if (t < 16) {
    const float4 bb = ((const float4*)pred_boxes)[rowBase + t];
    const float x0 = bb.x - 0.5f * bb.z, y0 = bb.y - 0.5f * bb.w;
    const float x1 = bb.x + 0.5f * bb.z, y1 = bb.y + 0.5f * bb.w;
    rowBox[t][0] = bb.x; rowBox[t][1] = bb.y; rowBox[t][2] = bb.z; rowBox[t][3] = bb.w;
    rowBox[t][4] = x0;   rowBox[t][5] = y0;   rowBox[t][6] = x1;   rowBox[t][7] = y1;
    rowBox[t][8] = (x1 - x0) * (y1 - y0);
  }
  // Stage target boxes + ids.
  if (t < 64) {
    const float4 bb = ((const float4*)tgt_boxes)[colBase + t];
    const float x0 = bb.x - 0.5f * bb.z, y0 = bb.y - 0.5f * bb.w;
    const float x1 = bb.x + 0.5f * bb.z, y1 = bb.y + 0.5f * bb.w;
    colBox[t][0] = bb.x; colBox[t][1] = bb.y; colBox[t][2] = bb.z; colBox[t][3] = bb.w;
    colBox[t][4] = x0;   colBox[t][5] = y0;   colBox[t][6] = x1;   colBox[t][7] = y1;
    colBox[t][8] = (x1 - x0) * (y1 - y0);
    colId[t] = tgt_labels[colBase + t];
  }
  __syncthreads();

  const int wave = t >> 5;
  const int lane = t & 31;
  const int hi   = lane >> 4;        // half-wave select
  const int l16  = lane & 15;
  const int colLocal = wave * 16 + l16;   // this lane's N (fixed for B & D)
  const int myTid = colId[colLocal];      // class id for this B column
  const int koff  = hi * 2;               // K sub-offset per A/B layout

  // D = P(16x92) x onehot(92x16), accumulated exactly in f32.
  v8f acc = {0.f, 0.f, 0.f, 0.f, 0.f, 0.f, 0.f, 0.f};
  #pragma unroll 4
  for (int kb = 0; kb < NCLS; kb += 4) {
    v2f a, b;
    a.x = probLds[l16 * 97 + kb + koff];
    a.y = probLds[l16 * 97 + kb + koff + 1];
    b.x = (myTid == kb + koff)     ? 1.0f : 0.0f;
    b.y = (myTid == kb + koff + 1) ? 1.0f : 0.0f;
    acc = __builtin_amdgcn_wmma_f32_16x16x4_f32(
        false, a, false, b, (short)0, acc, false, false);
  }

  // Epilogue: acc[v] = prob[m][tgt_id[n]] for m = v + 8*hi, n = colLocal.
  const float bcx = colBox[colLocal][0], bcy = colBox[colLocal][1];
  const float bw  = colBox[colLocal][2], bh  = colBox[colLocal][3];
  const float bx0 = colBox[colLocal][4], by0 = colBox[colLocal][5];
  const float bx1 = colBox[colLocal][6], by1 = colBox[colLocal][7];
  const float bA  = colBox[colLocal][8];

  #pragma unroll
  for (int v = 0; v < 8; ++v) {
    const int m = v + 8 * hi;
    const float acx = rowBox[m][0], acy = rowBox[m][1];
    const float aw  = rowBox[m][2], ah  = rowBox[m][3];
    const float ax0 = rowBox[m][4], ay0 = rowBox[m][5];
    const float ax1 = rowBox[m][6], ay1 = rowBox[m][7];
    const float aA  = rowBox[m][8];

    const float l1 = fabsf(acx - bcx) + fabsf(acy - bcy) +
                     fabsf(aw - bw)   + fabsf(ah - bh);

    const float ix0 = fmaxf(ax0, bx0), iy0 = fmaxf(ay0, by0);
    const float ix1 = fminf(ax1, bx1), iy1 = fminf(ay1, by1);
    const float iw  = fmaxf(ix1 - ix0, 0.0f), ih = fmaxf(iy1 - iy0, 0.0f);
    const float inter = iw * ih;
    const float uni   = aA + bA - inter;
    const float iou   = inter / uni;

    const float ex0 = fminf(ax0, bx0), ey0 = fminf(ay0, by0);
    const float ex1 = fmaxf(ax1, bx1), ey1 = fmaxf(ay1, by1);
    const float ew  = fmaxf(ex1 - ex0, 0.0f), eh = fmaxf(ey1 - ey0, 0.0f);
    const float areaE = ew * eh;
    const float giou  = iou - (areaE - uni) / areaE;

    const float cost = 5.0f * l1 - acc[v] - 2.0f * giou;
    out[(long)(rowBase + m) * NCOLS + (colBase + colLocal)] = cost;
  }
}

extern "C" void kernel_launch(void* const* d_in, const int* in_sizes, int n_in,
                              void* d_out, int out_size, void* d_ws, size_t ws_size,
                              hipStream_t stream) {
  (void)in_sizes; (void)n_in; (void)out_size; (void)ws_size;
  const float* pred_logits = (const float*)d_in[0];  // (32,300,92)
  const float* pred_boxes  = (const float*)d_in[1];  // (32,300,4)
  const int*   tgt_labels  = (const int*)d_in[2];    // (32,50)
  const float* tgt_boxes   = (const float*)d_in[3];  // (32,50,4)
  float* out = (float*)d_out;                        // (32,300,1600)
  float* prob = (float*)d_ws;                        // (9600,92) scratch

  // Softmax: 9600 rows, one wave per row, 8 waves per block.
  hm_softmax_rows<<<NROWS / 8, 256, 0, stream>>>(pred_logits, prob);

  // Cost matrix: 600 row-tiles x 25 col-groups, 128 threads (4 waves).
  dim3 grid(NROWS / 16, NCOLS / 64);
  hm_cost_kernel<<<grid, 128, 0, stream>>>(prob, pred_boxes, tgt_labels,
                                           tgt_boxes, out);
}